// GraphConvFeatureExtractorV2_88510686036731
// MI455X (gfx1250) — compile-verified
//
#include <hip/hip_runtime.h>

#define NNODES 50000
#define NEDGES 800000
#define HD 64
#define OUTD 128
#define NL 4
#define AST 68  // padded LDS stride (floats) for 16xK A tiles: conflict-free A-frag loads

typedef float v2f __attribute__((ext_vector_type(2)));
typedef float v8f __attribute__((ext_vector_type(8)));

// ---- B staged in fragment-native layout: float4 per (k-group, col) = B[4kg..4kg+3][n].
// Lane reads its (B[k][n], B[k+1][n]) pair as ONE aligned ds_load_b64 -> even VGPR pair,
// no v_mov marshalling. Banks: lanes0-15 words 4n..4n+1, lanes16-31 words 4n+2..4n+3:
// all 64 LDS banks, conflict-free.
template <int BS>
__device__ __forceinline__ void stage_Bfrag(const float* __restrict__ G, float4* sB4,
                                            int tid) {
  for (int i = tid; i < HD * BS; i += 256) {
    const int k = i / BS;
    const int n = i % BS;
    ((float*)&sB4[(k >> 2) * BS + n])[k & 3] = G[i];
  }
}

// Wave-level fp32 GEMM: C(16 x NT*16) += A(16x64 LDS, stride AST) * B(64 x NT*16, frag LDS)
// via V_WMMA_F32_16X16X4_F32, 16-step K loop.
template <int NT, int BS>
__device__ __forceinline__ void wave_gemm(const float* __restrict__ A,
                                          const float4* __restrict__ Bf, v8f* acc) {
  const int lane = threadIdx.x & 31;
  const int mn = lane & 15;
  const int kh = lane >> 4;       // 0 or 1
  const int ko = kh << 1;         // 0 or 2
#pragma unroll
  for (int k0 = 0; k0 < HD; k0 += 4) {
    const int kg = k0 >> 2;
    const float2 aa = *(const float2*)&A[mn * AST + k0 + ko];
    v2f a;
    a.x = aa.x;
    a.y = aa.y;
#pragma unroll
    for (int nt = 0; nt < NT; ++nt) {
      const float2 bb = ((const float2*)&Bf[kg * BS + nt * 16 + mn])[kh];
      v2f b;
      b.x = bb.x;
      b.y = bb.y;
      acc[nt] = __builtin_amdgcn_wmma_f32_16x16x4_f32(
          false, a, false, b, (short)0, acc[nt], false, false);
    }
  }
}

// ---------------- Kernel 1: edge_attr = relu(rel_pos @ ew1 + eb1) @ ew2 + eb2 ----------------
__global__ __launch_bounds__(256) void edge_attr_kernel(
    const float* __restrict__ x, const int* __restrict__ eidx,
    const float* __restrict__ ew1, const float* __restrict__ eb1,
    const float* __restrict__ ew2, const float* __restrict__ eb2,
    float* __restrict__ edge_attr) {
  __shared__ float sA[128 * AST];
  __shared__ float4 sB4[(HD / 4) * HD];
  __shared__ float sb2[HD];
  const int tid = threadIdx.x;
  const int e0 = blockIdx.x * 128;
  stage_Bfrag<HD>(ew2, sB4, tid);
  if (tid < HD) sb2[tid] = eb2[tid];
  {  // hidden layer (2 -> 64) in VALU; thread t does edge t/2, 32 cols
    const int le = tid >> 1;
    const int c0 = (tid & 1) * 32;
    const int e = e0 + le;
    float rx = 0.f, ry = 0.f;
    if (e < NEDGES) {
      const int s = eidx[e];
      const int d = eidx[NEDGES + e];
      rx = x[d * 3 + 0] - x[s * 3 + 0];
      ry = x[d * 3 + 1] - x[s * 3 + 1];
    }
#pragma unroll
    for (int j = 0; j < 32; ++j) {
      const int jj = c0 + j;
      sA[le * AST + jj] = fmaxf(rx * ew1[jj] + ry * ew1[HD + jj] + eb1[jj], 0.f);
    }
  }
  __syncthreads();
  const int wave = tid >> 5;
  const int lane = tid & 31;
  const int mn = lane & 15;
  const int rowb = (lane >> 4) * 8;
  v8f acc[4] = {};
  wave_gemm<4, HD>(&sA[wave * 16 * AST], sB4, acc);
#pragma unroll
  for (int nt = 0; nt < 4; ++nt)
#pragma unroll
    for (int r = 0; r < 8; ++r) {
      const int e = e0 + wave * 16 + rowb + r;
      if (e < NEDGES)
        edge_attr[(size_t)e * HD + nt * 16 + mn] = acc[nt][r] + sb2[nt * 16 + mn];
    }
}

// ---------------- Kernel 2: h = x @ in_w + in_b (3 -> 64, VALU) ----------------
__global__ __launch_bounds__(256) void input_proj_kernel(
    const float* __restrict__ x, const float* __restrict__ in_w,
    const float* __restrict__ in_b, float* __restrict__ h) {
  const unsigned gid = blockIdx.x * 256u + threadIdx.x;
  if (gid >= (unsigned)NNODES * 16u) return;
  const int node = gid >> 4;
  const int c0 = (gid & 15) << 2;
  const float x0 = x[node * 3 + 0], x1 = x[node * 3 + 1], x2 = x[node * 3 + 2];
  float4 o;
  float* po = &o.x;
#pragma unroll
  for (int j = 0; j < 4; ++j)
    po[j] = x0 * in_w[c0 + j] + x1 * in_w[HD + c0 + j] + x2 * in_w[2 * HD + c0 + j] +
            in_b[c0 + j];
  *(float4*)(h + (size_t)node * HD + c0) = o;
}

// ---------------- Kernel 3: clear agg ----------------
__global__ __launch_bounds__(256) void clear_kernel(float* __restrict__ p, int n4) {
  const unsigned gid = blockIdx.x * 256u + threadIdx.x;
  if (gid < (unsigned)n4)
    *(float4*)(p + (size_t)gid * 4) = make_float4(0.f, 0.f, 0.f, 0.f);
}

// ---------------- Kernel 4: msg = relu(h[row] + ea); agg[col] += msg ----------------
__global__ __launch_bounds__(256) void message_agg_kernel(
    const float* __restrict__ h, const float* __restrict__ edge_attr,
    const int* __restrict__ eidx, float* __restrict__ agg) {
  const unsigned gid = blockIdx.x * 256u + threadIdx.x;
  if (gid >= (unsigned)NEDGES * 16u) return;
  const unsigned e = gid >> 4;
  const unsigned c = (gid & 15u) << 2;
  const int src = eidx[e];
  const int dst = eidx[NEDGES + e];
  const float4 hv = *(const float4*)(h + (size_t)src * HD + c);        // L2-resident
  const float4 ea = *(const float4*)(edge_attr + (size_t)e * HD + c);  // HBM stream
  float4 m;
  m.x = fmaxf(hv.x + ea.x, 0.f);
  m.y = fmaxf(hv.y + ea.y, 0.f);
  m.z = fmaxf(hv.z + ea.z, 0.f);
  m.w = fmaxf(hv.w + ea.w, 0.f);
  float* dp = agg + (size_t)dst * HD + c;
  __hip_atomic_fetch_add(dp + 0, m.x, __ATOMIC_RELAXED, __HIP_MEMORY_SCOPE_AGENT);
  __hip_atomic_fetch_add(dp + 1, m.y, __ATOMIC_RELAXED, __HIP_MEMORY_SCOPE_AGENT);
  __hip_atomic_fetch_add(dp + 2, m.z, __ATOMIC_RELAXED, __HIP_MEMORY_SCOPE_AGENT);
  __hip_atomic_fetch_add(dp + 3, m.w, __ATOMIC_RELAXED, __HIP_MEMORY_SCOPE_AGENT);
}

// ---------------- Kernel 5: fused node update: 2x GEMM(WMMA) + LN + GELU + residual ----------------
__global__ __launch_bounds__(256) void node_update_kernel(
    float* __restrict__ h, const float* __restrict__ agg,
    const float* __restrict__ cw1, const float* __restrict__ cb1,
    const float* __restrict__ cw2, const float* __restrict__ cb2,
    const float* __restrict__ lnw, const float* __restrict__ lnb) {
  __shared__ float sA[128 * AST];
  __shared__ float4 sB1[(HD / 4) * HD];
  __shared__ float4 sB2[(HD / 4) * HD];
  __shared__ float sbias[HD * 4];
  const int tid = threadIdx.x;
  const int n0 = blockIdx.x * 128;
  stage_Bfrag<HD>(cw1, sB1, tid);
  stage_Bfrag<HD>(cw2, sB2, tid);
  if (tid < HD) {
    sbias[tid] = cb1[tid];
    sbias[HD + tid] = cb2[tid];
    sbias[2 * HD + tid] = lnw[tid];
    sbias[3 * HD + tid] = lnb[tid];
  }
  for (int i = tid; i < 128 * 16; i += 256) {  // stage m = h + agg
    const int ln = i >> 4;
    const int c = (i & 15) << 2;
    const int node = n0 + ln;
    float4 v = make_float4(0.f, 0.f, 0.f, 0.f);
    if (node < NNODES) {
      const float4 hv = *(const float4*)(h + (size_t)node * HD + c);
      const float4 av = *(const float4*)(agg + (size_t)node * HD + c);
      v.x = hv.x + av.x;
      v.y = hv.y + av.y;
      v.z = hv.z + av.z;
      v.w = hv.w + av.w;
    }
    *(float4*)(&sA[ln * AST + c]) = v;
  }
  __syncthreads();
  const int wave = tid >> 5;
  const int lane = tid & 31;
  const int mn = lane & 15;
  const int rowb = (lane >> 4) * 8;
  float* Aw = &sA[wave * 16 * AST];
  v8f acc[4] = {};
  wave_gemm<4, HD>(Aw, sB1, acc);
  // bias + ReLU, write back into this wave's private A slice (wave-local, DS in-order)
#pragma unroll
  for (int nt = 0; nt < 4; ++nt)
#pragma unroll
    for (int r = 0; r < 8; ++r)
      Aw[(rowb + r) * AST + nt * 16 + mn] = fmaxf(acc[nt][r] + sbias[nt * 16 + mn], 0.f);
  v8f acc2[4] = {};
  wave_gemm<4, HD>(Aw, sB2, acc2);
#pragma unroll
  for (int nt = 0; nt < 4; ++nt)
#pragma unroll
    for (int r = 0; r < 8; ++r) acc2[nt][r] += sbias[HD + nt * 16 + mn];
  // LayerNorm over 64 cols on the C-tile register layout (row = one 16-lane half).
  float mu[8], rs[8];
#pragma unroll
  for (int r = 0; r < 8; ++r) {
    float s = acc2[0][r] + acc2[1][r] + acc2[2][r] + acc2[3][r];
    s += __shfl_xor(s, 1, 32);
    s += __shfl_xor(s, 2, 32);
    s += __shfl_xor(s, 4, 32);
    s += __shfl_xor(s, 8, 32);
    mu[r] = s * (1.f / 64.f);
    float v = 0.f;
#pragma unroll
    for (int nt = 0; nt < 4; ++nt) {
      const float d = acc2[nt][r] - mu[r];
      v += d * d;
    }
    v += __shfl_xor(v, 1, 32);
    v += __shfl_xor(v, 2, 32);
    v += __shfl_xor(v, 4, 32);
    v += __shfl_xor(v, 8, 32);
    rs[r] = rsqrtf(v * (1.f / 64.f) + 1e-5f);
  }
#pragma unroll
  for (int r = 0; r < 8; ++r) {
    const int node = n0 + wave * 16 + rowb + r;
    if (node < NNODES) {
#pragma unroll
      for (int nt = 0; nt < 4; ++nt) {
        const int col = nt * 16 + mn;
        const float t =
            (acc2[nt][r] - mu[r]) * rs[r] * sbias[2 * HD + col] + sbias[3 * HD + col];
        const float g = 0.5f * t * (1.f + erff(t * 0.70710678118654752f));  // exact GELU
        h[(size_t)node * HD + col] = g + h[(size_t)node * HD + col];        // + identity
      }
    }
  }
}

// ---------------- Kernel 6: out = LN(h @ (fw+rw) + (fb+rb)) ----------------
__global__ __launch_bounds__(256) void output_kernel(
    const float* __restrict__ h, const float* __restrict__ fw,
    const float* __restrict__ fb, const float* __restrict__ rw,
    const float* __restrict__ rb, const float* __restrict__ ow,
    const float* __restrict__ ob, float* __restrict__ out) {
  __shared__ float sA[128 * AST];
  __shared__ float4 sB4[(HD / 4) * OUTD];
  __shared__ float sbias[OUTD * 3];
  const int tid = threadIdx.x;
  const int n0 = blockIdx.x * 128;
  for (int i = tid; i < HD * OUTD; i += 256) {  // fold two GEMMs: W = fw + rw
    const int k = i >> 7;
    const int n = i & (OUTD - 1);
    ((float*)&sB4[(k >> 2) * OUTD + n])[k & 3] = fw[i] + rw[i];
  }
  if (tid < OUTD) {
    sbias[tid] = fb[tid] + rb[tid];
    sbias[OUTD + tid] = ow[tid];
    sbias[2 * OUTD + tid] = ob[tid];
  }
  for (int i = tid; i < 128 * 16; i += 256) {
    const int ln = i >> 4;
    const int c = (i & 15) << 2;
    const int node = n0 + ln;
    float4 v = make_float4(0.f, 0.f, 0.f, 0.f);
    if (node < NNODES) v = *(const float4*)(h + (size_t)node * HD + c);
    *(float4*)(&sA[ln * AST + c]) = v;
  }
  __syncthreads();
  const int wave = tid >> 5;
  const int lane = tid & 31;
  const int mn = lane & 15;
  const int rowb = (lane >> 4) * 8;
  v8f acc[8] = {};
  wave_gemm<8, OUTD>(&sA[wave * 16 * AST], sB4, acc);
#pragma unroll
  for (int nt = 0; nt < 8; ++nt)
#pragma unroll
    for (int r = 0; r < 8; ++r) acc[nt][r] += sbias[nt * 16 + mn];
  float mu[8], rs[8];
#pragma unroll
  for (int r = 0; r < 8; ++r) {
    float s = 0.f;
#pragma unroll
    for (int nt = 0; nt < 8; ++nt) s += acc[nt][r];
    s += __shfl_xor(s, 1, 32);
    s += __shfl_xor(s, 2, 32);
    s += __shfl_xor(s, 4, 32);
    s += __shfl_xor(s, 8, 32);
    mu[r] = s * (1.f / 128.f);
    float v = 0.f;
#pragma unroll
    for (int nt = 0; nt < 8; ++nt) {
      const float d = acc[nt][r] - mu[r];
      v += d * d;
    }
    v += __shfl_xor(v, 1, 32);
    v += __shfl_xor(v, 2, 32);
    v += __shfl_xor(v, 4, 32);
    v += __shfl_xor(v, 8, 32);
    rs[r] = rsqrtf(v * (1.f / 128.f) + 1e-5f);
  }
#pragma unroll
  for (int r = 0; r < 8; ++r) {
    const int node = n0 + wave * 16 + rowb + r;
    if (node < NNODES) {
#pragma unroll
      for (int nt = 0; nt < 8; ++nt) {
        const int col = nt * 16 + mn;
        out[(size_t)node * OUTD + col] =
            (acc[nt][r] - mu[r]) * rs[r] * sbias[OUTD + col] + sbias[2 * OUTD + col];
      }
    }
  }
}

extern "C" void kernel_launch(void* const* d_in, const int* in_sizes, int n_in,
                              void* d_out, int out_size, void* d_ws, size_t ws_size,
                              hipStream_t stream) {
  const float* x = (const float*)d_in[0];
  const int* eidx = (const int*)d_in[1];
  const float* in_w = (const float*)d_in[2];
  const float* in_b = (const float*)d_in[3];
  const float* ew1 = (const float*)d_in[4];
  const float* eb1 = (const float*)d_in[5];
  const float* ew2 = (const float*)d_in[6];
  const float* eb2 = (const float*)d_in[7];
  const float* cw1 = (const float*)d_in[8];
  const float* cb1 = (const float*)d_in[9];
  const float* cw2 = (const float*)d_in[10];
  const float* cb2 = (const float*)d_in[11];
  const float* lnw = (const float*)d_in[12];
  const float* lnb = (const float*)d_in[13];
  const float* fw = (const float*)d_in[14];
  const float* fb = (const float*)d_in[15];
  const float* rw = (const float*)d_in[16];
  const float* rb = (const float*)d_in[17];
  const float* ow = (const float*)d_in[18];
  const float* ob = (const float*)d_in[19];
  float* out = (float*)d_out;

  float* edge_attr = (float*)d_ws;             // E*64 f32 (204.8 MB)
  float* h = edge_attr + (size_t)NEDGES * HD;  // N*64 f32 (12.8 MB, L2-resident)
  float* agg = h + (size_t)NNODES * HD;        // N*64 f32 (12.8 MB, L2-resident)

  edge_attr_kernel<<<(NEDGES + 127) / 128, 256, 0, stream>>>(x, eidx, ew1, eb1, ew2,
                                                             eb2, edge_attr);
  input_proj_kernel<<<(NNODES * 16 + 255) / 256, 256, 0, stream>>>(x, in_w, in_b, h);
  for (int l = 0; l < NL; ++l) {
    clear_kernel<<<(NNODES * HD / 4 + 255) / 256, 256, 0, stream>>>(agg,
                                                                    NNODES * HD / 4);
    message_agg_kernel<<<(NEDGES * 16 + 255) / 256, 256, 0, stream>>>(h, edge_attr,
                                                                      eidx, agg);
    node_update_kernel<<<(NNODES + 127) / 128, 256, 0, stream>>>(
        h, agg, cw1 + (size_t)l * HD * HD, cb1 + (size_t)l * HD,
        cw2 + (size_t)l * HD * HD, cb2 + (size_t)l * HD, lnw + (size_t)l * HD,
        lnb + (size_t)l * HD);
  }
  output_kernel<<<(NNODES + 127) / 128, 256, 0, stream>>>(h, fw, fb, rw, rb, ow, ob,
                                                          out);
}